// CenterLoss_51127290691819
// MI455X (gfx1250) — compile-verified
//
#include <hip/hip_runtime.h>

typedef float v2f __attribute__((ext_vector_type(2)));
typedef float v8f __attribute__((ext_vector_type(8)));

#define BATCH        512
#define DIMS         2048
#define TILE_M       16
#define NUM_TILES    (BATCH / TILE_M)        // 32 blocks
#define WAVES        16                      // waves per block (512 threads)
#define K_PER_WAVE   (DIMS / WAVES)          // 128
#define WMMA_K       4
#define K_ITERS      (K_PER_WAVE / WMMA_K)   // 32 wmma steps per wave

// One block = one 16-sample tile. Wave w reduces K-chunk [w*128, w*128+128)
// via the Gram-diagonal trick: D = Diff * Diff^T accumulated with
// V_WMMA_F32_16X16X4_F32; diag(D) are the squared distances.
__global__ __launch_bounds__(WAVES * 32) void center_loss_wmma(
    const float* __restrict__ features,
    const float* __restrict__ centers,
    const int*   __restrict__ targets,
    float* __restrict__ out)
{
    __shared__ float partial[WAVES * TILE_M];

    const int tid  = (int)threadIdx.x;
    const int wave = tid >> 5;
    const int lane = tid & 31;
    const int m    = lane & 15;                    // row of the 16-sample tile
    const int sample = (int)blockIdx.x * TILE_M + m;

    const int tgt   = targets[sample];
    // A-matrix f32 16x4 layout: lanes 0-15 hold K = {0,1}, lanes 16-31 K = {2,3}.
    // B (4x16, N across lanes, same K split) is bit-identical for Gram products.
    const int kbase = wave * K_PER_WAVE + ((lane >> 4) << 1);

    const float* __restrict__ fp = features + (size_t)sample * DIMS + kbase;
    const float* __restrict__ cp = centers  + (size_t)tgt    * DIMS + kbase;

    v8f acc = {};
#pragma unroll 4
    for (int it = 0; it < K_ITERS; ++it) {
        v2f f = *(const v2f*)(fp + it * WMMA_K);
        v2f c = *(const v2f*)(cp + it * WMMA_K);
        v2f d = f - c;
        // (neg_a, A, neg_b, B, c_mod, C, reuse_a, reuse_b)
        acc = __builtin_amdgcn_wmma_f32_16x16x4_f32(
            false, d, false, d, (short)0, acc, false, false);
    }

    // Diagonal of 16x16 f32 D: D[r,r] -> VGPR r lane r (r<8), VGPR r-8 lane 16+r (r>=8)
    const int sel = (lane < 8) ? lane : (lane - 24);
    float v = acc[0];
    v = (sel == 1) ? acc[1] : v;
    v = (sel == 2) ? acc[2] : v;
    v = (sel == 3) ? acc[3] : v;
    v = (sel == 4) ? acc[4] : v;
    v = (sel == 5) ? acc[5] : v;
    v = (sel == 6) ? acc[6] : v;
    v = (sel == 7) ? acc[7] : v;
    if (lane < 8 || lane >= 24)
        partial[wave * TILE_M + m] = v;
    __syncthreads();

    // Cross-wave K reduction, then clip per sample, then mean contribution.
    if (tid < TILE_M) {
        float dist = 0.0f;
#pragma unroll
        for (int w = 0; w < WAVES; ++w)
            dist += partial[w * TILE_M + tid];
        dist = fminf(fmaxf(dist, 1e-12f), 1e12f);
        atomicAdd(out, dist * (1.0f / (float)BATCH));
    }
}

__global__ void center_loss_zero(float* __restrict__ out)
{
    if (threadIdx.x == 0) out[0] = 0.0f;
}

extern "C" void kernel_launch(void* const* d_in, const int* in_sizes, int n_in,
                              void* d_out, int out_size, void* d_ws, size_t ws_size,
                              hipStream_t stream)
{
    (void)in_sizes; (void)n_in; (void)d_ws; (void)ws_size; (void)out_size;
    const float* features = (const float*)d_in[0];
    const float* centers  = (const float*)d_in[1];
    const int*   targets  = (const int*)d_in[2];
    float* out = (float*)d_out;

    center_loss_zero<<<1, 32, 0, stream>>>(out);
    center_loss_wmma<<<NUM_TILES, WAVES * 32, 0, stream>>>(features, centers, targets, out);
}